// Interaction_MM_layer_9698036155143
// MI455X (gfx1250) — compile-verified
//
#include <hip/hip_runtime.h>
#include <hip/hip_bf16.h>

// ---------------------------------------------------------------------------
// Interaction_MM_layer forward for MI455X (gfx1250, wave32).
//
// Problem sizes: BS=4, N_NODES=48, FEAT=64, HID=64, E=2256.
// Total ~1.5 GFLOP over ~5 MB -> latency bound on MI455X (23.3 TB/s HBM,
// 192 MB L2). Strategy: 4 kernels, LDS-resident intermediates, wave-level
// SSM scans, f32 WMMA (16x16x4) for the node-MMS dense projections (the only
// GEMM-shaped work; f32 WMMA keeps reference precision).
// ---------------------------------------------------------------------------

#define NN    48
#define FEATD 64
#define BSZ   4
#define NEDGE (NN * (NN - 1))   // 2256

typedef __attribute__((ext_vector_type(2))) float v2f;
typedef __attribute__((ext_vector_type(8))) float v8f;

struct MMSParams {
  const float* in_w;   // (2*d, d)
  const float* conv_w; // (d, 2)
  const float* conv_b; // (d,)
  const float* x_w;    // (dt_rank + 2*64, d)
  const float* dt_w;   // (d, dt_rank)
  const float* dt_b;   // (d,)
  const float* A_log;  // (d, 64)
  const float* Dp;     // (d,)
  const float* out_w;  // (d, d)
  const float* dim_w;  // (n_out, d)
  const float* dim_b;  // (n_out,)
  const float* ln_g;   // ln shape
  const float* ln_b;   // ln shape
};

__device__ __forceinline__ float sigm(float x)      { return 1.0f / (1.0f + __expf(-x)); }
__device__ __forceinline__ float siluf(float x)     { return x * sigm(x); }
__device__ __forceinline__ float softplusf(float x) { return (x > 20.0f) ? x : log1pf(__expf(x)); }

__device__ __forceinline__ float wred32(float v) {
#pragma unroll
  for (int m = 16; m >= 1; m >>= 1) v += __shfl_xor(v, m, 32);
  return v;
}

// ---------------------------------------------------------------------------
// f32 WMMA tile GEMM:  Dst[M x N] = A[M x K] * W^T, with W given [N x K]
// row-major (weights as stored by the reference: y = x @ W.T).
// Fragment layouts per CDNA5 ISA 7.12.2:
//   A (16x4 f32): lanes 0-15 -> M=lane, VGPR{0,1}=K{0,1}; lanes 16-31 -> K{2,3}.
//   B (4x16 f32): assumed mirror of A (lane half selects K base 0/2, vector
//                 element selects +0/+1, N = lane%16). Unverifiable w/o HW.
//   C/D (16x16):  VGPR r -> M = r + 8*(lane/16), N = lane%16.
// M, N must be multiples of 16; K multiple of 4. All 32 lanes active (EXEC=~0).
// ---------------------------------------------------------------------------
__device__ inline void gemm_wmma_f32(const float* __restrict__ A, int lda,
                                     const float* __restrict__ W, int ldw,
                                     float* __restrict__ Dst, int ldd,
                                     int Mt, int Nt, int K) {
  const int lid  = threadIdx.x & 31;
  const int wv   = threadIdx.x >> 5;
  const int nw   = blockDim.x >> 5;
  const int half = lid >> 4;
  const int sub  = lid & 15;
  for (int tile = wv; tile < Mt * Nt; tile += nw) {
    const int m0 = (tile / Nt) * 16;
    const int n0 = (tile % Nt) * 16;
    v8f acc = {0.f, 0.f, 0.f, 0.f, 0.f, 0.f, 0.f, 0.f};
#if __has_builtin(__builtin_amdgcn_wmma_f32_16x16x4_f32)
    for (int k0 = 0; k0 < K; k0 += 4) {
      const int ka = k0 + 2 * half;
      v2f a, b;
      a.x = A[(m0 + sub) * lda + ka];
      a.y = A[(m0 + sub) * lda + ka + 1];
      b.x = W[(n0 + sub) * ldw + ka];
      b.y = W[(n0 + sub) * ldw + ka + 1];
      acc = __builtin_amdgcn_wmma_f32_16x16x4_f32(false, a, false, b,
                                                  (short)0, acc, false, false);
    }
#else
#warning "f32 WMMA 16x16x4 builtin unavailable on this toolchain; scalar fallback in use"
    for (int k0 = 0; k0 < K; ++k0) {
#pragma unroll
      for (int r = 0; r < 8; ++r)
        acc[r] += A[(m0 + r + 8 * half) * lda + k0] * W[(n0 + sub) * ldw + k0];
    }
#endif
#pragma unroll
    for (int r = 0; r < 8; ++r)
      Dst[(m0 + r + 8 * half) * ldd + n0 + sub] = acc[r];
  }
}

// ---------------------------------------------------------------------------
// Zero-init (workspace is poisoned by the harness, not re-zeroed per call).
// ---------------------------------------------------------------------------
__global__ void zero_kernel(float* __restrict__ p, int n) {
  int i = blockIdx.x * blockDim.x + threadIdx.x;
  if (i < n) p[i] = 0.0f;
}

// ---------------------------------------------------------------------------
// MMS with d_model=2, L=64 (edge MMS / node-edge MMS). One wave32 per
// sequence; 8 waves per 256-thread block. Lanes act over L in elementwise
// phases (l = lane, lane+32) and over d_state (s = lane, lane+32) in the scan.
// MODE 0: edge sequences -> writes mij = mms*mask^2, atomicAdd segment-sum.
// MODE 1: node-edge sequences -> writes out = mms*mask.
// ---------------------------------------------------------------------------
template <int MODE>
__global__ __launch_bounds__(256) void mms2_kernel(
    const float* __restrict__ nf,        // node_feature (BS,NN,FEAT)
    const float* __restrict__ chan1,     // MODE1: incoming_mms (BS,NN,FEAT)
    const int*   __restrict__ send_idx,
    const int*   __restrict__ recv_idx,
    const float* __restrict__ mask,      // edges_mask (BS,E) / nodes_mask (BS,NN)
    MMSParams p,
    float* __restrict__ outbuf,          // MODE0: mij (BS,E,FEAT); MODE1: out (BS,NN,FEAT)
    float* __restrict__ incoming,        // MODE0: (BS,NN,FEAT) atomic accumulator
    int total)                           // number of sequences
{
  __shared__ float sXp[8][FEATD][2];  // pre-conv x
  __shared__ float sXc[8][FEATD][2];  // post conv+silu x
  __shared__ float sZ [8][FEATD][2];
  __shared__ float sDt[8][FEATD][2];
  __shared__ float sYv[8][FEATD][2];

  const int lid = threadIdx.x & 31;
  const int wv  = threadIdx.x >> 5;
  int seq = blockIdx.x * 8 + wv;
  const bool valid = (seq < total);
  if (!valid) seq = 0;

  int b, rcv = 0, srcA, srcB;
  float maskv;
  if (MODE == 0) {
    b = seq / NEDGE;
    const int e = seq - b * NEDGE;
    const int snd = send_idx[e];
    rcv = recv_idx[e];
    srcA = (b * NN + snd) * FEATD;   // channel 0 = sender
    srcB = (b * NN + rcv) * FEATD;   // channel 1 = receiver
    maskv = mask[b * NEDGE + e];
  } else {
    b = seq / NN;
    const int nd = seq - b * NN;
    srcA = (b * NN + nd) * FEATD;    // channel 0 = node_feature
    srcB = (b * NN + nd) * FEATD;    // channel 1 = incoming (from chan1)
    maskv = mask[b * NN + nd];
  }

  // small parameter loads (wave-uniform -> scalar loads)
  const float iw00 = p.in_w[0], iw01 = p.in_w[1];
  const float iw10 = p.in_w[2], iw11 = p.in_w[3];
  const float iw20 = p.in_w[4], iw21 = p.in_w[5];
  const float iw30 = p.in_w[6], iw31 = p.in_w[7];
  const float cw00 = p.conv_w[0], cw01 = p.conv_w[1];
  const float cw10 = p.conv_w[2], cw11 = p.conv_w[3];
  const float cb0 = p.conv_b[0], cb1 = p.conv_b[1];
  const float xw0 = p.x_w[0], xw1 = p.x_w[1];          // x_w row 0 (dt_rank=1)
  const float dtw0 = p.dt_w[0], dtw1 = p.dt_w[1];
  const float dtb0 = p.dt_b[0], dtb1 = p.dt_b[1];
  const float Dp0 = p.Dp[0], Dp1 = p.Dp[1];
  const float ow00 = p.out_w[0], ow01 = p.out_w[1];
  const float ow10 = p.out_w[2], ow11 = p.out_w[3];
  const float dw0 = p.dim_w[0], dw1 = p.dim_w[1];
  const float db  = p.dim_b[0];

  // ---- Phase A: in-projection (per l) ----
#pragma unroll
  for (int i = 0; i < 2; ++i) {
    const int l = lid + 32 * i;
    const float u0 = nf[srcA + l];
    const float u1 = (MODE == 0) ? nf[srcB + l] : chan1[srcB + l];
    sXp[wv][l][0] = u0 * iw00 + u1 * iw01;
    sXp[wv][l][1] = u0 * iw10 + u1 * iw11;
    sZ [wv][l][0] = u0 * iw20 + u1 * iw21;
    sZ [wv][l][1] = u0 * iw30 + u1 * iw31;
  }
  __syncthreads();

  // ---- Phase B: causal conv(k=2) + silu, dt projection ----
#pragma unroll
  for (int i = 0; i < 2; ++i) {
    const int l = lid + 32 * i;
    const float pm0 = (l > 0) ? sXp[wv][l - 1][0] : 0.0f;
    const float pm1 = (l > 0) ? sXp[wv][l - 1][1] : 0.0f;
    const float x0 = siluf(pm0 * cw00 + sXp[wv][l][0] * cw01 + cb0);
    const float x1 = siluf(pm1 * cw10 + sXp[wv][l][1] * cw11 + cb1);
    const float dtr = x0 * xw0 + x1 * xw1;
    sXc[wv][l][0] = x0;
    sXc[wv][l][1] = x1;
    sDt[wv][l][0] = softplusf(dtr * dtw0 + dtb0);
    sDt[wv][l][1] = softplusf(dtr * dtw1 + dtb1);
  }
  __syncthreads();

  // ---- Phase C: selective scan; lanes hold s = lid and s = lid+32 ----
  const int s0 = lid, s1 = lid + 32;
  const float A00 = -__expf(p.A_log[s0]),      A01 = -__expf(p.A_log[s1]);
  const float A10 = -__expf(p.A_log[64 + s0]), A11 = -__expf(p.A_log[64 + s1]);
  const float wB0x = p.x_w[(1 + s0) * 2],  wB0y = p.x_w[(1 + s0) * 2 + 1];
  const float wB1x = p.x_w[(1 + s1) * 2],  wB1y = p.x_w[(1 + s1) * 2 + 1];
  const float wC0x = p.x_w[(65 + s0) * 2], wC0y = p.x_w[(65 + s0) * 2 + 1];
  const float wC1x = p.x_w[(65 + s1) * 2], wC1y = p.x_w[(65 + s1) * 2 + 1];
  float h00 = 0.f, h01 = 0.f, h10 = 0.f, h11 = 0.f;
  for (int l = 0; l < FEATD; ++l) {
    const float x0 = sXc[wv][l][0], x1 = sXc[wv][l][1];
    const float d0 = sDt[wv][l][0], d1 = sDt[wv][l][1];
    const float Bs0 = x0 * wB0x + x1 * wB0y;
    const float Bs1 = x0 * wB1x + x1 * wB1y;
    const float Cs0 = x0 * wC0x + x1 * wC0y;
    const float Cs1 = x0 * wC1x + x1 * wC1y;
    h00 = __expf(d0 * A00) * h00 + (d0 * x0) * Bs0;
    h01 = __expf(d0 * A01) * h01 + (d0 * x0) * Bs1;
    h10 = __expf(d1 * A10) * h10 + (d1 * x1) * Bs0;
    h11 = __expf(d1 * A11) * h11 + (d1 * x1) * Bs1;
    float p0 = h00 * Cs0 + h01 * Cs1;
    float p1 = h10 * Cs0 + h11 * Cs1;
    p0 = wred32(p0);
    p1 = wred32(p1);
    if (lid == 0) { sYv[wv][l][0] = p0; sYv[wv][l][1] = p1; }
  }
  __syncthreads();

  // ---- Phase D: gates, out/dim projections, layernorm over 64 values ----
  float vloc[2];
  float ssum = 0.f, ssq = 0.f;
#pragma unroll
  for (int i = 0; i < 2; ++i) {
    const int l = lid + 32 * i;
    float y0 = sYv[wv][l][0] + sXc[wv][l][0] * Dp0;
    float y1 = sYv[wv][l][1] + sXc[wv][l][1] * Dp1;
    y0 *= siluf(sZ[wv][l][0]);
    y1 *= siluf(sZ[wv][l][1]);
    const float o0 = y0 * ow00 + y1 * ow01;
    const float o1 = y0 * ow10 + y1 * ow11;
    const float q0 = siluf(o0), q1 = siluf(o1);
    const float v = q0 * dw0 + q1 * dw1 + db;
    vloc[i] = v;
    ssum += v;
    ssq += v * v;
  }
  ssum = wred32(ssum);
  ssq  = wred32(ssq);
  const float mu = ssum * (1.0f / 64.0f);
  const float var = ssq * (1.0f / 64.0f) - mu * mu;
  const float rs = rsqrtf(var + 1e-5f);

#pragma unroll
  for (int i = 0; i < 2; ++i) {
    const int l = lid + 32 * i;
    const float msg = (vloc[i] - mu) * rs * p.ln_g[l] + p.ln_b[l];
    if (valid) {
      if (MODE == 0) {
        const float mv = msg * maskv * maskv;           // mij = mms * mask^2
        outbuf[seq * FEATD + l] = mv;
        atomicAdd(&incoming[(b * NN + rcv) * FEATD + l], mv);
      } else {
        outbuf[seq * FEATD + l] = msg * maskv;
      }
    }
  }
}

// ---------------------------------------------------------------------------
// Node MMS: d_model=64, L=48, one block (256 threads) per batch element.
// All intermediates LDS-resident; projections via f32 WMMA; scan keeps the
// 64x64 state as 16 regs/thread (thread t -> d = t/4, s-quarter = t%4).
// ---------------------------------------------------------------------------
#define NSM_U   0
#define NSM_XZ  (NSM_U  + NN * 64)        // in-proj out: [x_pre | z], ld 128
#define NSM_X   (NSM_XZ + NN * 128)       // post conv+silu x
#define NSM_DT  (NSM_X  + NN * 64)
#define NSM_BC  (NSM_DT + NN * 64)        // [B | C], ld 128
#define NSM_Y   (NSM_BC + NN * 128)
#define NSM_O   (NSM_Y  + NN * 64)
#define NSM_F   (NSM_O  + NN * 64)
#define NSM_DTR (NSM_F  + NN * 64)        // dt_raw (48 x 4)
#define NSM_RED (NSM_DTR + NN * 4)        // 16 reduction slots
#define NSM_TOTAL (NSM_RED + 16)

__global__ __launch_bounds__(256) void node_mms_kernel(
    const float* __restrict__ incoming,   // (BS,NN,64) segment sums
    MMSParams p,
    float* __restrict__ outbuf)           // (BS,NN,64) incoming_mms
{
  extern __shared__ float sm[];
  float* sU  = sm + NSM_U;
  float* sXZ = sm + NSM_XZ;
  float* sX  = sm + NSM_X;
  float* sDT = sm + NSM_DT;
  float* sBC = sm + NSM_BC;
  float* sY  = sm + NSM_Y;
  float* sO  = sm + NSM_O;
  float* sF  = sm + NSM_F;
  float* sDtr = sm + NSM_DTR;
  float* sRed = sm + NSM_RED;

  const int b = blockIdx.x;
  const int t = threadIdx.x;
  const int lid = t & 31;
  const int wv = t >> 5;

  // u = incoming / n
  for (int idx = t; idx < NN * 64; idx += 256)
    sU[idx] = incoming[b * NN * 64 + idx] * (1.0f / (float)NN);
  __syncthreads();

  // in-projection: (48x64) @ in_w.T(64x128) -> [x_pre | z]
  gemm_wmma_f32(sU, 64, p.in_w, 64, sXZ, 128, 3, 8, 64);
  __syncthreads();

  // causal conv (k=2) + silu over L per channel
  for (int idx = t; idx < NN * 64; idx += 256) {
    const int l = idx >> 6, d = idx & 63;
    const float xc = sXZ[l * 128 + d];
    const float xp = (l > 0) ? sXZ[(l - 1) * 128 + d] : 0.0f;
    sX[idx] = siluf(xp * p.conv_w[d * 2] + xc * p.conv_w[d * 2 + 1] + p.conv_b[d]);
  }
  __syncthreads();

  // dt_raw = x @ x_w[0:4].T  (48 x 4)
  if (t < NN * 4) {
    const int l = t >> 2, r = t & 3;
    float acc = 0.f;
    for (int k = 0; k < 64; ++k) acc += sX[l * 64 + k] * p.x_w[r * 64 + k];
    sDtr[t] = acc;
  }
  __syncthreads();

  // dt = softplus(dt_raw @ dt_w.T + dt_b)  (48 x 64)
  for (int idx = t; idx < NN * 64; idx += 256) {
    const int l = idx >> 6, d = idx & 63;
    float v = p.dt_b[d];
#pragma unroll
    for (int r = 0; r < 4; ++r) v += sDtr[l * 4 + r] * p.dt_w[d * 4 + r];
    sDT[idx] = softplusf(v);
  }
  // B,C = x @ x_w[4:132].T -> sBC (48 x 128), cols 0-63 = B, 64-127 = C
  gemm_wmma_f32(sX, 64, p.x_w + 4 * 64, 64, sBC, 128, 3, 8, 64);
  __syncthreads();

  // selective scan: h(64x64) distributed as 16 regs/thread
  {
    const int d = t >> 2, q = t & 3;
    float h[16], Ad[16];
#pragma unroll
    for (int i = 0; i < 16; ++i) {
      const int s = q * 16 + i;
      Ad[i] = -__expf(p.A_log[d * 64 + s]);
      h[i] = 0.0f;
    }
    for (int l = 0; l < NN; ++l) {
      const float dtv = sDT[l * 64 + d];
      const float xv  = sX[l * 64 + d];
      const float dx  = dtv * xv;
      float part = 0.0f;
#pragma unroll
      for (int i = 0; i < 16; ++i) {
        const int s = q * 16 + i;
        h[i] = __expf(dtv * Ad[i]) * h[i] + dx * sBC[l * 128 + s];
        part += h[i] * sBC[l * 128 + 64 + s];
      }
      part += __shfl_xor(part, 1, 32);
      part += __shfl_xor(part, 2, 32);
      if (q == 0) sY[l * 64 + d] = part;
    }
  }
  __syncthreads();

  // y = (y + x*D) * silu(z)
  for (int idx = t; idx < NN * 64; idx += 256) {
    const int l = idx >> 6, d = idx & 63;
    const float y = sY[idx] + sX[idx] * p.Dp[d];
    sY[idx] = y * siluf(sXZ[l * 128 + 64 + d]);
  }
  __syncthreads();

  // out = y @ out_w.T
  gemm_wmma_f32(sY, 64, p.out_w, 64, sO, 64, 3, 4, 64);
  __syncthreads();
  for (int idx = t; idx < NN * 64; idx += 256) sO[idx] = siluf(sO[idx]);
  __syncthreads();

  // f = silu(out) @ dim_w.T + dim_b
  gemm_wmma_f32(sO, 64, p.dim_w, 64, sF, 64, 3, 4, 64);
  __syncthreads();

  // layernorm over all (48 x 64) values
  float ls = 0.f, lq = 0.f;
  for (int idx = t; idx < NN * 64; idx += 256) {
    const float v = sF[idx] + p.dim_b[idx & 63];
    sF[idx] = v;
    ls += v;
    lq += v * v;
  }
  ls = wred32(ls);
  lq = wred32(lq);
  if (lid == 0) { sRed[wv] = ls; sRed[8 + wv] = lq; }
  __syncthreads();
  if (t == 0) {
    float s = 0.f, qq = 0.f;
    for (int w = 0; w < 8; ++w) { s += sRed[w]; qq += sRed[8 + w]; }
    const float mu = s / (float)(NN * 64);
    const float var = qq / (float)(NN * 64) - mu * mu;
    sRed[0] = mu;
    sRed[1] = rsqrtf(var + 1e-5f);
  }
  __syncthreads();
  const float mu = sRed[0], rs = sRed[1];
  for (int idx = t; idx < NN * 64; idx += 256)
    outbuf[b * NN * 64 + idx] = (sF[idx] - mu) * rs * p.ln_g[idx] + p.ln_b[idx];
}

// ---------------------------------------------------------------------------
// Host side. Input order = setup_inputs() insertion order with nested param
// dicts flattened in their insertion order:
//   0 node_feature, 1 t_nodes, 2 t_edges, 3 nodes_mask, 4 edges_mask,
//   5..17 edge_params {in_w,conv_w,conv_b,x_w,dt_w,dt_b,A_log,D,out_w,
//                      dim_w,dim_b,ln_g,ln_b},
//   18..30 node_params (same keys), 31..43 ne_params (same keys),
//   44 recv_idx, 45 send_idx.
// Output: out (BS*NN*FEAT) then mij (BS*E*FEAT), concatenated.
// ---------------------------------------------------------------------------
static MMSParams make_params(void* const* d_in, int base) {
  MMSParams p;
  p.in_w   = (const float*)d_in[base + 0];
  p.conv_w = (const float*)d_in[base + 1];
  p.conv_b = (const float*)d_in[base + 2];
  p.x_w    = (const float*)d_in[base + 3];
  p.dt_w   = (const float*)d_in[base + 4];
  p.dt_b   = (const float*)d_in[base + 5];
  p.A_log  = (const float*)d_in[base + 6];
  p.Dp     = (const float*)d_in[base + 7];
  p.out_w  = (const float*)d_in[base + 8];
  p.dim_w  = (const float*)d_in[base + 9];
  p.dim_b  = (const float*)d_in[base + 10];
  p.ln_g   = (const float*)d_in[base + 11];
  p.ln_b   = (const float*)d_in[base + 12];
  return p;
}

extern "C" void kernel_launch(void* const* d_in, const int* in_sizes, int n_in,
                              void* d_out, int out_size, void* d_ws, size_t ws_size,
                              hipStream_t stream) {
  const float* node_feature = (const float*)d_in[0];
  const float* nodes_mask   = (const float*)d_in[3];
  const float* edges_mask   = (const float*)d_in[4];
  MMSParams edge_p = make_params(d_in, 5);
  MMSParams node_p = make_params(d_in, 18);
  MMSParams ne_p   = make_params(d_in, 31);
  const int* recv_idx = (const int*)d_in[44];
  const int* send_idx = (const int*)d_in[45];

  float* out = (float*)d_out;                       // (BS,NN,FEAT)
  float* mij = out + BSZ * NN * FEATD;              // (BS,E,FEAT)

  float* incoming = (float*)d_ws;                   // (BS,NN,FEAT) accumulator
  float* inc_mms  = incoming + BSZ * NN * FEATD;    // (BS,NN,FEAT) node-MMS out

  const int nInc = BSZ * NN * FEATD;
  zero_kernel<<<(nInc + 255) / 256, 256, 0, stream>>>(incoming, nInc);

  // Edge MMS: 9024 sequences, 8 waves/block -> 1128 blocks (exact).
  const int nEdgeSeq = BSZ * NEDGE;
  mms2_kernel<0><<<nEdgeSeq / 8, 256, 0, stream>>>(
      node_feature, node_feature, send_idx, recv_idx, edges_mask,
      edge_p, mij, incoming, nEdgeSeq);

  // Node MMS: 1 block per batch element, 121 KB dynamic LDS (<= 320 KB/WGP).
  const size_t smemBytes = (size_t)NSM_TOTAL * sizeof(float);
  node_mms_kernel<<<BSZ, 256, smemBytes, stream>>>(incoming, node_p, inc_mms);

  // Node-edge MMS: 192 sequences -> 24 blocks (exact).
  const int nNodeSeq = BSZ * NN;
  mms2_kernel<1><<<nNodeSeq / 8, 256, 0, stream>>>(
      node_feature, inc_mms, send_idx, recv_idx, nodes_mask,
      ne_p, out, nullptr, nNodeSeq);
}